// YukselSpline_19018115187077
// MI455X (gfx1250) — compile-verified
//
#include <hip/hip_runtime.h>
#include <cstdint>

typedef __attribute__((ext_vector_type(4))) unsigned int v4u;
typedef __attribute__((ext_vector_type(8))) int          v8i;
typedef __attribute__((ext_vector_type(4))) int          v4i;
typedef __attribute__((ext_vector_type(4))) float        v4f;

#define NPTS 33
#define NSEG 30            // S-1, S = NPTS-2
#define ROW  36            // dwords per segment row: 30 used + 6 pad.
                           // 36i mod 64 -> 16 distinct 16B-aligned bank chunks
                           // => <=2-way LDS conflicts, b128-loadable rows.
#define TABF (NSEG * ROW)  // 1080 floats = 4320 bytes

// ---------------------------------------------------------------------------
// Tiny setup kernel: unroll the jax.lax.scan carry (depends only on P) into a
// flat coefficient table C[i*ROW + k]:
//   [0..5]  Pn  (3x2 row-major)   [6..11]  Pn2 (3x2)
//   [12..20] Sn (3x3 row-major)   [21..29] Sn2 (3x3)   [30..35] pad
// ---------------------------------------------------------------------------
__global__ __launch_bounds__(64) void yuksel_setup(const float* __restrict__ P,
                                                   float* __restrict__ C) {
  if (threadIdx.x != 0) return;
  float p[NPTS][2], sig[NPTS][3];
  for (int j = 0; j < NPTS; ++j) {
    p[j][0] = P[j * 5 + 0];
    p[j][1] = P[j * 5 + 1];
    float a = P[j * 5 + 2], b = P[j * 5 + 3], c = P[j * 5 + 4];
    sig[j][0] = a * a + b * b;
    sig[j][1] = b * (a + c);
    sig[j][2] = b * b + c * c;
  }
  float Pn[3][2], Sn[3][3];
  for (int r = 0; r < 3; ++r) {
    Pn[r][0] = p[r][0]; Pn[r][1] = p[r][1];
    Sn[r][0] = sig[r][0]; Sn[r][1] = sig[r][1]; Sn[r][2] = sig[r][2];
  }
  for (int i = 0; i < NSEG; ++i) {
    float* c = C + i * ROW;
    float P2[3][2], S2[3][3];
    for (int k = 0; k < 2; ++k) {
      float p0 = (Pn[0][k] + Pn[2][k]) * 0.25f + Pn[1][k] * 0.5f;
      float p2 = p[i + 3][k];
      float p1 = 2.0f * (Pn[2][k] - (p0 + p2) * 0.25f);
      P2[0][k] = p0; P2[1][k] = p1; P2[2][k] = p2;
    }
    for (int k = 0; k < 3; ++k) {
      float s0 = (Sn[0][k] + Sn[2][k]) * 0.0625f + Sn[1][k] * 0.25f;
      float s2 = sig[i + 3][k];
      float s1 = 4.0f * (Sn[2][k] - (s0 + s2) * 0.0625f);
      S2[0][k] = s0; S2[1][k] = s1; S2[2][k] = s2;
    }
    for (int r = 0; r < 3; ++r)
      for (int k = 0; k < 2; ++k) { c[r * 2 + k] = Pn[r][k]; c[6 + r * 2 + k] = P2[r][k]; }
    for (int r = 0; r < 3; ++r)
      for (int k = 0; k < 3; ++k) { c[12 + r * 3 + k] = Sn[r][k]; c[21 + r * 3 + k] = S2[r][k]; }
    for (int k = 30; k < ROW; ++k) c[k] = 0.0f;
    for (int r = 0; r < 3; ++r) {
      Pn[r][0] = P2[r][0]; Pn[r][1] = P2[r][1];
      Sn[r][0] = S2[r][0]; Sn[r][1] = S2[r][1]; Sn[r][2] = S2[r][2];
    }
  }
}

// ---------------------------------------------------------------------------
// Per-point evaluation against the LDS table.
// ---------------------------------------------------------------------------
__device__ __forceinline__ void eval_point(float t, const float* __restrict__ tab,
                                           float* __restrict__ m, float* __restrict__ s) {
  float fi = t * (float)NSEG;
  int i = (int)fi;                       // t >= 0 -> trunc == floor
  i = i < 0 ? 0 : (i > NSEG - 1 ? NSEG - 1 : i);
  float d = (fi - (float)i) * 0.5f;      // d in [0, 0.5]
  float sp, cp;
  sincospif(d, &sp, &cp);
  float cw = cp * cp, sw = sp * sp;
  float u = d + 0.5f, v = 0.5f - d;      // basis(d+0.5): (1-u)=0.5-d
  // Window weight folded into the basis once; serves mu linearly and sg via
  // squares, since cw^2 * b^2 == (cw*b)^2.
  float wp0 = cw * (v * v), wp1 = cw * (2.0f * u * v), wp2 = cw * (u * u);
  float e = 1.0f - d;                    // basis(d)
  float wc0 = sw * (e * e), wc1 = sw * (2.0f * d * e), wc2 = sw * (d * d);
  const float* c = tab + i * ROW;
  m[0] = wp0 * c[0] + wp1 * c[2] + wp2 * c[4] +
         wc0 * c[6] + wc1 * c[8] + wc2 * c[10];
  m[1] = wp0 * c[1] + wp1 * c[3] + wp2 * c[5] +
         wc0 * c[7] + wc1 * c[9] + wc2 * c[11];
  float qp0 = wp0 * wp0, qp1 = wp1 * wp1, qp2 = wp2 * wp2;
  float qc0 = wc0 * wc0, qc1 = wc1 * wc1, qc2 = wc2 * wc2;
  s[0] = qp0 * c[12] + qp1 * c[15] + qp2 * c[18] +
         qc0 * c[21] + qc1 * c[24] + qc2 * c[27];
  s[1] = qp0 * c[13] + qp1 * c[16] + qp2 * c[19] +
         qc0 * c[22] + qc1 * c[25] + qc2 * c[28];
  s[2] = qp0 * c[14] + qp1 * c[17] + qp2 * c[20] +
         qc0 * c[23] + qc1 * c[26] + qc2 * c[29];
}

// ---------------------------------------------------------------------------
// Main streaming kernel: 4 points per thread, 128-bit NT loads/stores.
// Coefficient table staged into LDS via the Tensor Data Mover (CDNA5 TDM).
// ---------------------------------------------------------------------------
__global__ __launch_bounds__(256) void yuksel_eval(const float* __restrict__ x,
                                                   const float* __restrict__ C,
                                                   float* __restrict__ out, int N) {
  __shared__ __attribute__((aligned(16))) float tab[TABF];

  // --- CDNA5 TDM: DMA the 4320-byte LUT global -> LDS (issued by wave 0) ---
  if (threadIdx.x < 32) {
    unsigned ldsa =
        (unsigned)__builtin_amdgcn_readfirstlane((int)(unsigned)(uintptr_t)(void*)tab);
    unsigned long long ga = (unsigned long long)(uintptr_t)C;
    // D# group0: count=1 valid, lds_addr, 57-bit global addr, type=2 ("image")
    v4u g0 = {1u, ldsa, (unsigned)(ga & 0xFFFFFFFFull),
              (unsigned)((ga >> 32) & 0x01FFFFFFull) | (2u << 30)};
    // D# group1: data_size=4B; tensor_dim0 = tile_dim0 = stride = 1080 dwords;
    // tensor_dim1 = tile_dim1 = 1; no multicast/pad/iterate/atomic-barrier.
    v8i g1 = {0x00020000,
              (int)((unsigned)TABF << 16),   // tensor_dim0[15:0]
              (int)(1u << 16),               // tensor_dim0 hi=0 | tensor_dim1=1
              (int)((unsigned)TABF << 16),   // tile_dim0
              1,                             // tile_dim1=1, tile_dim2=0
              TABF,                          // tensor_dim0_stride
              0, 0};
    v4i z4 = {0, 0, 0, 0};
    v8i z8 = {0, 0, 0, 0, 0, 0, 0, 0};
    __builtin_amdgcn_tensor_load_to_lds(g0, g1, z4, z4, z8, 0);
    __builtin_amdgcn_s_wait_tensorcnt(0);
  }
  __syncthreads();
  // Belt-and-braces: overwrite with plain loads so numerical correctness does
  // not hinge on the hand-built descriptor encoding above.
  for (int k = threadIdx.x; k < TABF; k += 256) tab[k] = C[k];
  __syncthreads();

  int g = blockIdx.x * blockDim.x + threadIdx.x;
  int base = g * 4;
  if (base >= N) return;
  float* sgout = out + 2 * (size_t)N;

  if (base + 4 <= N && (N & 3) == 0) {
    v4f xv = __builtin_nontemporal_load((const v4f*)x + g);
    float m[4][2], s[4][3];
    eval_point(xv.x, tab, m[0], s[0]);
    eval_point(xv.y, tab, m[1], s[1]);
    eval_point(xv.z, tab, m[2], s[2]);
    eval_point(xv.w, tab, m[3], s[3]);
    v4f mu0 = {m[0][0], m[0][1], m[1][0], m[1][1]};
    v4f mu1 = {m[2][0], m[2][1], m[3][0], m[3][1]};
    __builtin_nontemporal_store(mu0, (v4f*)out + 2 * (size_t)g);
    __builtin_nontemporal_store(mu1, (v4f*)out + 2 * (size_t)g + 1);
    v4f sg0 = {s[0][0], s[0][1], s[0][2], s[1][0]};
    v4f sg1 = {s[1][1], s[1][2], s[2][0], s[2][1]};
    v4f sg2 = {s[2][2], s[3][0], s[3][1], s[3][2]};
    __builtin_nontemporal_store(sg0, (v4f*)sgout + 3 * (size_t)g);
    __builtin_nontemporal_store(sg1, (v4f*)sgout + 3 * (size_t)g + 1);
    __builtin_nontemporal_store(sg2, (v4f*)sgout + 3 * (size_t)g + 2);
  } else {
    for (int n = base; n < N && n < base + 4; ++n) {
      float m[2], s[3];
      eval_point(x[n], tab, m, s);
      out[2 * (size_t)n]     = m[0];
      out[2 * (size_t)n + 1] = m[1];
      sgout[3 * (size_t)n]     = s[0];
      sgout[3 * (size_t)n + 1] = s[1];
      sgout[3 * (size_t)n + 2] = s[2];
    }
  }
}

extern "C" void kernel_launch(void* const* d_in, const int* in_sizes, int n_in,
                              void* d_out, int out_size, void* d_ws, size_t ws_size,
                              hipStream_t stream) {
  (void)n_in; (void)out_size; (void)ws_size;
  const float* x = (const float*)d_in[0];
  const float* P = (const float*)d_in[1];
  float* out = (float*)d_out;
  float* C   = (float*)d_ws;   // 4320-byte coefficient table
  int N = in_sizes[0];

  yuksel_setup<<<1, 64, 0, stream>>>(P, C);

  int nthreads = (N + 3) / 4;
  int block = 256;
  int grid = (nthreads + block - 1) / block;
  yuksel_eval<<<grid, block, 0, stream>>>(x, C, out, N);
}